// JointLoss_42314017800264
// MI455X (gfx1250) — compile-verified
//
#include <hip/hip_runtime.h>
#include <hip/hip_bf16.h>
#include <math.h>

// ---------------------------------------------------------------------------
// JointLoss for MI455X (gfx1250). Bandwidth-bound streaming pass (311 MB of
// f32 inputs -> ~13.4 us at 23.3 TB/s) + WMMA-based cross-lane reduction of
// the 13 per-workgroup accumulators + tiny per-batch eigen epilogue.
// ---------------------------------------------------------------------------

#define EPSV 1e-8f
#define B_   32
#define H_   288
#define W_   384
#define HW_  (H_ * W_)
#define BLOCKS_PER_BATCH 64
#define THREADS 256
#define NSLOT 16   // 0-5 Hm, 6-8 g, 9 n_num, 10 mask_sum, 11 grad_num, 12 grad_cnt

typedef __attribute__((ext_vector_type(2))) float v2f;
typedef __attribute__((ext_vector_type(8))) float v8f;

__device__ __forceinline__ void load_unit3(const float* p, size_t idx, float* o) {
    float x = p[idx], y = p[idx + (size_t)HW_], z = p[idx + 2 * (size_t)HW_];
    float n = sqrtf(x * x + y * y + z * z) + EPSV;
    o[0] = x / n; o[1] = y / n; o[2] = z / n;
}

// ---------------------------------------------------------------------------
// Phase 1: stream all pixels of one batch per blockIdx.y, accumulate 13 slots
// per lane, reduce across the 256-thread workgroup with LDS + chained
// v_wmma_f32_16x16x4_f32 (A = ones, columns = slots), atomicAdd per batch.
// ---------------------------------------------------------------------------
__global__ __launch_bounds__(THREADS) void joint_loss_phase1(
    const float* __restrict__ cam,   // [B,9,H,W]
    const float* __restrict__ up,    // [B,3,H,W]
    const float* __restrict__ wts,   // [B,3,H,W]
    const float* __restrict__ pn,    // [B,3,H,W]
    const float* __restrict__ gn,    // [B,3,H,W]
    const float* __restrict__ mask,  // [B,H,W]
    float* __restrict__ accbuf)      // [B,16]
{
    const int b   = blockIdx.y;
    const int tid = threadIdx.x;

    float acc[NSLOT];
#pragma unroll
    for (int s = 0; s < NSLOT; ++s) acc[s] = 0.0f;

    const size_t mbase0 = (size_t)b * HW_;
    const size_t nbase0 = (size_t)b * 3 * HW_;
    const size_t cbase0 = (size_t)b * 9 * HW_;

    for (int p = blockIdx.x * THREADS + tid; p < HW_; p += BLOCKS_PER_BATCH * THREADS) {
        const int h = p / W_;
        const int w = p - h * W_;

        const float m = mask[mbase0 + p];
        const size_t nb = nbase0 + p;

        // ---- cosine-similarity term ----
        float gu[3], pu[3];
        load_unit3(gn, nb, gu);
        load_unit3(pn, nb, pu);
        float dot = gu[0] * pu[0] + gu[1] * pu[1] + gu[2] * pu[2];
        float na  = sqrtf(gu[0] * gu[0] + gu[1] * gu[1] + gu[2] * gu[2]);
        float nbn = sqrtf(pu[0] * pu[0] + pu[1] * pu[1] + pu[2] * pu[2]);
        float cosv = dot / fmaxf(na * nbn, EPSV);
        acc[9]  += m * (1.0f - cosv);
        acc[10] += m;

        const float d0x = pu[0] - gu[0], d0y = pu[1] - gu[1], d0z = pu[2] - gu[2];

        // ---- gradient term (vertical pair h, h+2) ----
        if (h < H_ - 2) {
            const size_t off = (size_t)(2 * W_);
            float mm = m * mask[mbase0 + p + off];
            float gu2[3], pu2[3];
            load_unit3(gn, nb + off, gu2);
            load_unit3(pn, nb + off, pu2);
            float s = fabsf(d0x - (pu2[0] - gu2[0])) +
                      fabsf(d0y - (pu2[1] - gu2[1])) +
                      fabsf(d0z - (pu2[2] - gu2[2]));
            acc[11] += mm * s;
            acc[12] += 3.0f * mm;
        }
        // ---- gradient term (horizontal pair w, w+2) ----
        if (w < W_ - 2) {
            float mm = m * mask[mbase0 + p + 2];
            float gu2[3], pu2[3];
            load_unit3(gn, nb + 2, gu2);
            load_unit3(pn, nb + 2, pu2);
            float s = fabsf(d0x - (pu2[0] - gu2[0])) +
                      fabsf(d0y - (pu2[1] - gu2[1])) +
                      fabsf(d0z - (pu2[2] - gu2[2]));
            acc[11] += mm * s;
            acc[12] += 3.0f * mm;
        }

        // ---- pose accumulation: Hm = sum w_i^2 u_i u_i^T, g = sum w_i^2 u_i up_i ----
        float upu[3];
        load_unit3(up, nb, upu);
        const size_t cb = cbase0 + p;
#pragma unroll
        for (int i = 0; i < 3; ++i) {
            float u[3];
            load_unit3(cam, cb + (size_t)(3 * i) * HW_, u);
            float wt = wts[nb + (size_t)i * HW_];
            float w2 = wt * wt;
            acc[0] += w2 * u[0] * u[0];
            acc[1] += w2 * u[0] * u[1];
            acc[2] += w2 * u[0] * u[2];
            acc[3] += w2 * u[1] * u[1];
            acc[4] += w2 * u[1] * u[2];
            acc[5] += w2 * u[2] * u[2];
            float sg = w2 * upu[i];
            acc[6] += sg * u[0];
            acc[7] += sg * u[1];
            acc[8] += sg * u[2];
        }
    }

    // ------------------------------------------------------------------
    // Workgroup reduction: LDS transpose staging, then wave 0 folds the 8
    // waves in VALU and reduces the 32 lanes with 8 chained WMMA f32 ops.
    // Column n of B carries slot n; A = ones(16x4) -> D columns = slot sums.
    // ------------------------------------------------------------------
    __shared__ float smem[THREADS][NSLOT + 1];
#pragma unroll
    for (int s = 0; s < NSLOT; ++s) smem[tid][s] = acc[s];
    __syncthreads();

    if ((tid >> 5) == 0) {            // wave 0 only (uniform branch, EXEC all-1s)
        const int lane = tid & 31;
        const int col  = lane & 15;
        const bool hi  = lane >= 16;

        v8f c = {0.f, 0.f, 0.f, 0.f, 0.f, 0.f, 0.f, 0.f};
        v2f a = {1.0f, 1.0f};         // A = all-ones 16x4

#pragma unroll
        for (int t = 0; t < 8; ++t) {
            const int j0 = 4 * t + (hi ? 2 : 0);
            const int j1 = j0 + 1;
            float b0 = 0.0f, b1 = 0.0f;
#pragma unroll
            for (int wv = 0; wv < 8; ++wv) {   // fold the 8 waves
                b0 += smem[wv * 32 + j0][col];
                b1 += smem[wv * 32 + j1][col];
            }
            v2f bm = {b0, b1};
            // D = ones(16x4) x B(4x16) + C  ->  column sums, accumulated over t
            c = __builtin_amdgcn_wmma_f32_16x16x4_f32(
                    false, a, false, bm, (short)0, c, false, false);
        }
        // every row of D is identical; c[0] in lane L holds slot (L%16) total
        if (lane < 13) atomicAdd(&accbuf[b * NSLOT + lane], c[0]);
    }
}

// ---------------------------------------------------------------------------
// Phase 2: one wave, one batch per lane. Smallest real root of
// det((H - l I)^2 - g g^T) = 0 (monic degree-6), then v = (H - l I)^-1 g,
// angle vs gt_up, and the final weighted sum of the three loss terms.
// ---------------------------------------------------------------------------
__device__ __forceinline__ void conv33(const float* a, const float* b, float* o) {
    o[0] = a[0] * b[0];
    o[1] = a[0] * b[1] + a[1] * b[0];
    o[2] = a[0] * b[2] + a[1] * b[1] + a[2] * b[0];
    o[3] = a[1] * b[2] + a[2] * b[1];
    o[4] = a[2] * b[2];
}

__device__ __forceinline__ float evalp6(const float* p, float x) {
    float v = p[6];
#pragma unroll
    for (int i = 5; i >= 0; --i) v = v * x + p[i];
    return v;
}

__global__ __launch_bounds__(32) void joint_loss_phase2(
    const float* __restrict__ accbuf,  // [B,16]
    const float* __restrict__ gt_up,   // [B,3]
    float* __restrict__ out)           // [1]
{
    const int b = threadIdx.x;  // 32 lanes == 32 batches

    const float h00 = accbuf[b * NSLOT + 0];
    const float h01 = accbuf[b * NSLOT + 1];
    const float h02 = accbuf[b * NSLOT + 2];
    const float h11 = accbuf[b * NSLOT + 3];
    const float h12 = accbuf[b * NSLOT + 4];
    const float h22 = accbuf[b * NSLOT + 5];
    const float g0  = accbuf[b * NSLOT + 6];
    const float g1  = accbuf[b * NSLOT + 7];
    const float g2  = accbuf[b * NSLOT + 8];

    // H^2 (symmetric)
    const float s00 = h00 * h00 + h01 * h01 + h02 * h02;
    const float s01 = h00 * h01 + h01 * h11 + h02 * h12;
    const float s02 = h00 * h02 + h01 * h12 + h02 * h22;
    const float s11 = h01 * h01 + h11 * h11 + h12 * h12;
    const float s12 = h01 * h02 + h11 * h12 + h12 * h22;
    const float s22 = h02 * h02 + h12 * h12 + h22 * h22;

    // M(l) = (H - l I)^2 - g g^T ; entries quadratic in l
    float e00[3] = {s00 - g0 * g0, -2.f * h00, 1.f};
    float e01[3] = {s01 - g0 * g1, -2.f * h01, 0.f};
    float e02[3] = {s02 - g0 * g2, -2.f * h02, 0.f};
    float e11[3] = {s11 - g1 * g1, -2.f * h11, 1.f};
    float e12[3] = {s12 - g1 * g2, -2.f * h12, 0.f};
    float e22[3] = {s22 - g2 * g2, -2.f * h22, 1.f};

    // det via symmetric cofactor expansion -> degree-6 polynomial p[0..6]
    float p[7];
#pragma unroll
    for (int i = 0; i < 7; ++i) p[i] = 0.f;
    float A1[5], A2[5], A3[5], tmp[5];
    conv33(e11, e22, A1); conv33(e12, e12, tmp);
#pragma unroll
    for (int i = 0; i < 5; ++i) A1[i] -= tmp[i];
    conv33(e01, e22, A2); conv33(e12, e02, tmp);
#pragma unroll
    for (int i = 0; i < 5; ++i) A2[i] -= tmp[i];
    conv33(e01, e12, A3); conv33(e11, e02, tmp);
#pragma unroll
    for (int i = 0; i < 5; ++i) A3[i] -= tmp[i];
#pragma unroll
    for (int i = 0; i < 3; ++i)
#pragma unroll
        for (int j = 0; j < 5; ++j)
            p[i + j] += e00[i] * A1[j] - e01[i] * A2[j] + e02[i] * A3[j];

    // normalize to monic (p[6] should be 1)
    float lead = p[6];
    if (fabsf(lead) < 1e-30f) lead = 1.f;
#pragma unroll
    for (int i = 0; i < 7; ++i) p[i] /= lead;

    // Cauchy bound, coarse scan for the smallest real root, then bisection
    float R = 0.f;
#pragma unroll
    for (int i = 0; i < 6; ++i) R = fmaxf(R, fabsf(p[i]));
    R = 1.f + R;

    const int NS = 4096;
    float lam = 0.f;
    bool found = false;
    float prevx = -R, prevv = evalp6(p, -R);
    float bestx = -R, bestabs = fabsf(prevv);
    for (int i = 1; i <= NS; ++i) {
        float x = -R + (2.f * R) * (float)i / (float)NS;
        float v = evalp6(p, x);
        float av = fabsf(v);
        if (av < bestabs) { bestabs = av; bestx = x; }
        if ((prevv <= 0.f) != (v <= 0.f)) {
            float lo = prevx, hi = x, flo = prevv;
            for (int it = 0; it < 64; ++it) {
                float mid = 0.5f * (lo + hi);
                float fm = evalp6(p, mid);
                if ((flo <= 0.f) == (fm <= 0.f)) { lo = mid; flo = fm; }
                else hi = mid;
            }
            lam = 0.5f * (lo + hi);
            found = true;
            break;
        }
        prevx = x; prevv = v;
    }
    if (!found) lam = bestx;

    // v = (H - lam I)^{-1} g via adjugate (symmetric)
    const float m00 = h00 - lam, m11 = h11 - lam, m22 = h22 - lam;
    const float a00 = m11 * m22 - h12 * h12;
    const float a01 = -(h01 * m22 - h12 * h02);
    const float a02 = h01 * h12 - m11 * h02;
    const float a11 = m00 * m22 - h02 * h02;
    const float a12 = -(m00 * h12 - h01 * h02);
    const float a22 = m00 * m11 - h01 * h01;
    float det = m00 * a00 + h01 * a01 + h02 * a02;
    if (fabsf(det) < 1e-30f) det = (det < 0.f) ? -1e-30f : 1e-30f;
    const float idet = 1.f / det;
    const float v0 = (a00 * g0 + a01 * g1 + a02 * g2) * idet;
    const float v1 = (a01 * g0 + a11 * g1 + a12 * g2) * idet;
    const float v2 = (a02 * g0 + a12 * g1 + a22 * g2) * idet;

    const float u0 = gt_up[b * 3 + 0];
    const float u1 = gt_up[b * 3 + 1];
    const float u2 = gt_up[b * 3 + 2];
    const float dvu = v0 * u0 + v1 * u1 + v2 * u2;
    const float nv = sqrtf(v0 * v0 + v1 * v1 + v2 * v2);
    const float nu = sqrtf(u0 * u0 + u1 * u1 + u2 * u2);
    const float cth = dvu / fmaxf(nv * nu, EPSV);
    float ang;
    if (cth < 1.f - 1e-6f) {
        float s = fminf(fmaxf(cth, -1.f), 1.f - 1e-6f);
        ang = acosf(s);
    } else {
        ang = 1.f - cth;
    }

    // global scalar terms + wave reduction (wave32)
    float n_num = accbuf[b * NSLOT + 9];
    float msum  = accbuf[b * NSLOT + 10];
    float gnum  = accbuf[b * NSLOT + 11];
    float gcnt  = accbuf[b * NSLOT + 12];
#pragma unroll
    for (int off = 16; off > 0; off >>= 1) {
        ang   += __shfl_down(ang,   off, 32);
        n_num += __shfl_down(n_num, off, 32);
        msum  += __shfl_down(msum,  off, 32);
        gnum  += __shfl_down(gnum,  off, 32);
        gcnt  += __shfl_down(gcnt,  off, 32);
    }
    if (b == 0) {
        const float n_term = n_num / (msum + EPSV);
        const float g_term = gnum / (gcnt + EPSV);
        const float pose   = ang / (float)B_;   // W_POSE * mean(ang)
        out[0] = 1.0f * n_term + 0.25f * g_term + 1.0f * pose;
    }
}

extern "C" void kernel_launch(void* const* d_in, const int* in_sizes, int n_in,
                              void* d_out, int out_size, void* d_ws, size_t ws_size,
                              hipStream_t stream) {
    const float* cam  = (const float*)d_in[0];  // pred_cam_geo [B,9,H,W]
    const float* up   = (const float*)d_in[1];  // pred_up_geo  [B,3,H,W]
    const float* wts  = (const float*)d_in[2];  // pred_weights [B,3,H,W]
    const float* pn   = (const float*)d_in[3];  // pred_n       [B,3,H,W]
    const float* gn   = (const float*)d_in[4];  // gt_n         [B,3,H,W]
    const float* mask = (const float*)d_in[5];  // mask         [B,H,W]
    const float* gup  = (const float*)d_in[6];  // gt_up_vector [B,3]

    float* accbuf = (float*)d_ws;               // [B,16] accumulators
    hipMemsetAsync(accbuf, 0, (size_t)B_ * NSLOT * sizeof(float), stream);

    dim3 grid(BLOCKS_PER_BATCH, B_);
    joint_loss_phase1<<<grid, THREADS, 0, stream>>>(cam, up, wts, pn, gn, mask, accbuf);
    joint_loss_phase2<<<1, 32, 0, stream>>>(accbuf, gup, (float*)d_out);
}